// LightStageField_54460185313787
// MI455X (gfx1250) — compile-verified
//
#include <hip/hip_runtime.h>
#include <hip/hip_bf16.h>
#include <math.h>

typedef __attribute__((ext_vector_type(2))) float v2f;
typedef __attribute__((ext_vector_type(8))) float v8f;

#define NLIGHT 256
#define HW     65536     // 256*256
#define CHW    196608    // 3*256*256

// Branchless sorted top-3 (ascending) insert: 3 compares + selects, no EXEC churn.
// Strict < keeps the lower index first on exact ties (matches stable top_k).
__device__ __forceinline__ void top3_insert(float s, int l,
    float& b0, float& b1, float& b2, int& j0, int& j1, int& j2) {
  const bool c0 = s < b0;
  const bool c1 = s < b1;
  const bool c2 = s < b2;
  const float nb2 = c1 ? b1 : (c2 ? s : b2);
  const int   nj2 = c1 ? j1 : (c2 ? l : j2);
  const float nb1 = c0 ? b0 : (c1 ? s : b1);
  const int   nj1 = c0 ? j0 : (c1 ? l : j1);
  const float nb0 = c0 ? s : b0;
  const int   nj0 = c0 ? l : j0;
  b0 = nb0; b1 = nb1; b2 = nb2;
  j0 = nj0; j1 = nj1; j2 = nj2;
}

__global__ __launch_bounds__(32) void LightStageField_kernel(
    const float* __restrict__ X,     // [B,4]
    const float* __restrict__ LS,    // [256,3,256,256]
    const float* __restrict__ LP,    // [256,2]
    float* __restrict__ out) {       // [B,3]
  // staging buffer: [16 lights][33] (32 points used, +1 pad -> conflict-free)
  __shared__ float sbuf[16 * 33];

  const int  lane = threadIdx.x;          // wave32
  const int  l16  = lane & 15;
  const bool hi   = lane >= 16;
  const long base = (long)blockIdx.x * 32; // 32 points per wave

  const float4* X4 = (const float4*)X;

  // Warm the cache for this lane's X row (consumed after the whole scan phase;
  // lanes 16-31 never touch it in the WMMA setup below).
  __builtin_prefetch(&X4[base + lane], 0, 0);

  // ---- A fragments for V_WMMA_F32_16X16X4_F32 (16x4 f32, 2 VGPRs/lane) ----
  // lanes 0-15 hold K=0,1 (= x2, x3); lanes 16-31 hold K=2,3 (= 1, 0)
  v2f A0, A1;
  if (!hi) {
    float4 xa = X4[base + l16];
    float4 xb = X4[base + 16 + l16];
    A0[0] = xa.z; A0[1] = xa.w;
    A1[0] = xb.z; A1[1] = xb.w;
  } else {
    A0[0] = 1.0f; A0[1] = 0.0f;
    A1[0] = 1.0f; A1[1] = 0.0f;
  }

  float b0 = 3.0e38f, b1 = 3.0e38f, b2 = 3.0e38f;
  int   j0 = 0, j1 = 0, j2 = 0;

  // ---- score[b,l] = |p_l|^2 - 2 x_b . p_l  (== d2 - |x|^2, same ordering) ----
  for (int t = 0; t < 16; ++t) {
    const int n0 = t << 4;
    const int l  = n0 + l16;
    const float p0 = LP[2 * l];
    const float p1 = LP[2 * l + 1];
    // B fragment (4x16 f32): lanes 0-15 hold K=0,1; lanes 16-31 hold K=2,3
    v2f Bf;
    if (!hi) { Bf[0] = -2.0f * p0; Bf[1] = -2.0f * p1; }
    else     { Bf[0] = p0 * p0 + p1 * p1; Bf[1] = 0.0f; }

    v8f C0 = {};
    v8f C1 = {};
    C0 = __builtin_amdgcn_wmma_f32_16x16x4_f32(false, A0, false, Bf, (short)0, C0, false, false);
    C1 = __builtin_amdgcn_wmma_f32_16x16x4_f32(false, A1, false, Bf, (short)0, C1, false, false);

    __syncthreads();  // previous tile's reads done before overwrite
    // D layout: VGPR r -> M = r (lanes 0-15) / r+8 (lanes 16-31), N = lane&15
    const int mb  = hi ? 8 : 0;
    const int row = l16 * 33;
    #pragma unroll
    for (int r = 0; r < 8; ++r) {
      sbuf[row + mb + r]      = C0[r];   // points 0..15
      sbuf[row + 16 + mb + r] = C1[r];   // points 16..31
    }
    __syncthreads();

    // Batch all 16 LDS reads first (pipelined ds_loads), then branchless inserts.
    float sv[16];
    #pragma unroll
    for (int q = 0; q < 16; ++q) sv[q] = sbuf[q * 33 + lane];
    #pragma unroll
    for (int q = 0; q < 16; ++q)
      top3_insert(sv[q], n0 + q, b0, b1, b2, j0, j1, j2);
  }

  // ---- per-lane point data ----
  const float4 xr  = X4[base + lane];
  const float  ptx = xr.z, pty = xr.w;

  // ---- barycentric weights (with outside/NaN fallback) ----
  const float c0x = LP[2 * j0], c0y = LP[2 * j0 + 1];
  const float c1x = LP[2 * j1], c1y = LP[2 * j1 + 1];
  const float c2x = LP[2 * j2], c2y = LP[2 * j2 + 1];
  const float v0x = c1x - c0x, v0y = c1y - c0y;
  const float v1x = c2x - c0x, v1y = c2y - c0y;
  const float v2x = ptx - c0x, v2y = pty - c0y;
  const float d00 = v0x * v0x + v0y * v0y;
  const float d11 = v1x * v1x + v1y * v1y;
  const float d01 = v0x * v1x + v0y * v1y;
  const float d20 = v2x * v0x + v2y * v0y;
  const float d21 = v2x * v1x + v2y * v1y;
  const float denom = d00 * d11 - d01 * d01;
  const float vb = (d11 * d20 - d01 * d21) / denom;
  const float wb = (d00 * d21 - d01 * d20) / denom;
  const float ub = 1.0f - vb - wb;
  const bool inside = (ub >= 0.0f) && (vb >= 0.0f) && (wb >= 0.0f); // NaN -> false
  float tt = d20 / d00;
  tt = (tt < 0.0f) ? 0.0f : ((tt > 1.0f) ? 1.0f : tt);  // NaN propagates (matches jnp.clip)
  const float w0 = inside ? ub : (1.0f - tt);
  const float w1 = inside ? vb : tt;
  const float w2 = inside ? wb : 0.0f;

  // ---- shared bilinear (x,y) footprint, align_corners + border ----
  const float px = fminf(fmaxf((xr.x + 1.0f) * 0.5f * 255.0f, 0.0f), 255.0f);
  const float py = fminf(fmaxf((xr.y + 1.0f) * 0.5f * 255.0f, 0.0f), 255.0f);
  const int   x0 = (int)floorf(px);
  const int   y0 = (int)floorf(py);
  const int   x1 = min(x0 + 1, 255);
  const int   y1 = min(y0 + 1, 255);
  const float wx = px - (float)x0;
  const float wy = py - (float)y0;
  const float w00 = (1.0f - wy) * (1.0f - wx);
  const float w01 = (1.0f - wy) * wx;
  const float w10 = wy * (1.0f - wx);
  const float w11 = wy * wx;
  const int o00 = y0 * 256 + x0, o01 = y0 * 256 + x1;
  const int o10 = y1 * 256 + x0, o11 = y1 * 256 + x1;

  float acc0 = 0.0f, acc1 = 0.0f, acc2 = 0.0f;
  const int   jidx[3] = { j0, j1, j2 };
  const float jwt[3]  = { w0, w1, w2 };

  #pragma unroll
  for (int j = 0; j < 3; ++j) {
    const int   l   = jidx[j];
    const float wgt = jwt[j];
    // exact reference fp sequence for z
    const float z  = (float)l / 127.5f - 1.0f;
    float pz = (z + 1.0f) * 0.5f * 255.0f;
    pz = fminf(fmaxf(pz, 0.0f), 255.0f);
    const int   z0 = (int)floorf(pz);
    const int   z1 = min(z0 + 1, 255);
    const float wz = pz - (float)z0;

    const float* s0 = LS + (size_t)z0 * CHW;
    float col0 = s0[o00] * w00 + s0[o01] * w01 + s0[o10] * w10 + s0[o11] * w11;
    float col1 = s0[HW + o00] * w00 + s0[HW + o01] * w01 + s0[HW + o10] * w10 + s0[HW + o11] * w11;
    float col2 = s0[2 * HW + o00] * w00 + s0[2 * HW + o01] * w01 + s0[2 * HW + o10] * w10 + s0[2 * HW + o11] * w11;

    if (wz != 0.0f) {  // z-interp collapses almost always; skip far slice
      const float* s1 = LS + (size_t)z1 * CHW;
      const float d0 = s1[o00] * w00 + s1[o01] * w01 + s1[o10] * w10 + s1[o11] * w11;
      const float d1 = s1[HW + o00] * w00 + s1[HW + o01] * w01 + s1[HW + o10] * w10 + s1[HW + o11] * w11;
      const float d2 = s1[2 * HW + o00] * w00 + s1[2 * HW + o01] * w01 + s1[2 * HW + o10] * w10 + s1[2 * HW + o11] * w11;
      col0 = col0 * (1.0f - wz) + d0 * wz;
      col1 = col1 * (1.0f - wz) + d1 * wz;
      col2 = col2 * (1.0f - wz) + d2 * wz;
    }
    acc0 += wgt * col0;
    acc1 += wgt * col1;
    acc2 += wgt * col2;
  }

  const long ob = (base + lane) * 3;
  out[ob + 0] = acc0;
  out[ob + 1] = acc1;
  out[ob + 2] = acc2;
}

extern "C" void kernel_launch(void* const* d_in, const int* in_sizes, int n_in,
                              void* d_out, int out_size, void* d_ws, size_t ws_size,
                              hipStream_t stream) {
  const float* X  = (const float*)d_in[0];   // [B,4] f32
  const float* LS = (const float*)d_in[1];   // [256,3,256,256] f32
  const float* LP = (const float*)d_in[2];   // [256,2] f32
  float* out = (float*)d_out;                // [B,3] f32

  const int B      = in_sizes[0] / 4;        // 262144
  const int blocks = B / 32;                 // 32 points per wave/block
  LightStageField_kernel<<<blocks, 32, 0, stream>>>(X, LS, LP, out);
}